// CEL_VAR_14998025798515
// MI455X (gfx1250) — compile-verified
//
#include <hip/hip_runtime.h>
#include <hip/hip_bf16.h>
#include <stdint.h>

// Problem constants (match reference)
#define S_SAMP 10
#define B_DIM  4
#define C_DIM  19
#define HW_DIM (512 * 512)
#define THREADS 256

// ---------------------------------------------------------------------------
// CDNA5 async DMA: global -> LDS, tracked with ASYNCcnt.
// GV mode: vdst = LDS byte offset (workgroup LDS address), vaddr = 64-bit global addr.
// Non-temporal hint: eps is streamed exactly once, keep it out of L2 residency.
// ---------------------------------------------------------------------------
__device__ __forceinline__ void async_load_f32_to_lds(uint32_t lds_off, uint64_t gaddr) {
    asm volatile("global_load_async_to_lds_b32 %0, %1, off th:TH_LOAD_NT"
                 :: "v"(lds_off), "v"(gaddr)
                 : "memory");
}

__device__ __forceinline__ void wait_async_le19() {
    asm volatile("s_wait_asynccnt 19" ::: "memory");
}
__device__ __forceinline__ void wait_async_le0() {
    asm volatile("s_wait_asynccnt 0" ::: "memory");
}

// ---------------------------------------------------------------------------
// Main kernel: one thread per pixel (b,h,w). Loads mean/std once into VGPRs,
// double-buffers eps per-sample through LDS via async DMA, accumulates NLL
// over the 10 MC samples, then does a deterministic block tree-reduce.
// ---------------------------------------------------------------------------
__global__ __launch_bounds__(THREADS) void cel_var_main_kernel(
    const float* __restrict__ mean,
    const float* __restrict__ logv,
    const int*   __restrict__ label,
    const float* __restrict__ eps,
    float*       __restrict__ partial)
{
    const int tid = threadIdx.x;
    const long long p  = (long long)blockIdx.x * THREADS + tid;   // pixel in [0, B*HW)
    const long long b  = p / HW_DIM;                              // block-uniform (256 | HW)
    const long long hw = p - b * (long long)HW_DIM;
    const long long base = b * (long long)C_DIM * HW_DIM + hw;    // index of (b, c=0, hw)

    // --- mean / std resident in VGPRs (read exactly once from HBM) ---------
    float m[C_DIM], sd[C_DIM];
#pragma unroll
    for (int c = 0; c < C_DIM; ++c) {
        m[c]  = __builtin_nontemporal_load(mean + base + (long long)c * HW_DIM);
        sd[c] = __expf(0.5f * __builtin_nontemporal_load(logv + base + (long long)c * HW_DIM));
    }
    const int lbl = label[p];

    // --- LDS double buffer for eps (stage = sample parity) -----------------
    __shared__ float lbuf[2][C_DIM * THREADS];           // 38,912 bytes
    const uint32_t lds_base   = (uint32_t)(uintptr_t)(&lbuf[0][0]);
    const uint64_t eps_base   = (uint64_t)(uintptr_t)eps + (uint64_t)base * 4ull;
    const uint64_t samp_bytes = (uint64_t)B_DIM * C_DIM * HW_DIM * 4ull;
    const uint64_t chan_bytes = (uint64_t)HW_DIM * 4ull;

    // Preload sample 0 into stage 0.
#pragma unroll
    for (int c = 0; c < C_DIM; ++c) {
        async_load_f32_to_lds(lds_base + (uint32_t)((c * THREADS + tid) * 4),
                              eps_base + (uint64_t)c * chan_bytes);
    }

    float acc = 0.0f;
    for (int s = 0; s < S_SAMP; ++s) {
        const int cur = s & 1;

        if (s + 1 < S_SAMP) {
            // Issue next stage's 19 async loads, then wait for the *older* 19
            // (async loads retire in order -> ASYNCcnt <= 19 means stage s done).
            const uint32_t doff  = lds_base + (uint32_t)((((cur ^ 1) * C_DIM) * THREADS + tid) * 4);
            const uint64_t gbase = eps_base + (uint64_t)(s + 1) * samp_bytes;
#pragma unroll
            for (int c = 0; c < C_DIM; ++c) {
                async_load_f32_to_lds(doff + (uint32_t)(c * THREADS * 4),
                                      gbase + (uint64_t)c * chan_bytes);
            }
            wait_async_le19();
        } else {
            wait_async_le0();
        }

        // Each lane reads back only the LDS slots it requested -> no barrier.
        volatile const float* ebuf = &lbuf[cur][0];

        float logits[C_DIM];
        float mx = -3.0e38f;
        float ll = 0.0f;
#pragma unroll
        for (int c = 0; c < C_DIM; ++c) {
            const float e  = ebuf[c * THREADS + tid];
            const float lg = fmaf(sd[c], e, m[c]);
            logits[c] = lg;
            mx = fmaxf(mx, lg);
            if (c == lbl) ll = lg;          // v_cndmask per unrolled c
        }
        float sum = 0.0f;
#pragma unroll
        for (int c = 0; c < C_DIM; ++c) sum += __expf(logits[c] - mx);

        // nll = -(logit[lbl] - mx - log(sum))
        acc += (mx + __logf(sum)) - ll;
    }

    // --- deterministic block tree reduction --------------------------------
    __shared__ float red[THREADS];
    red[tid] = acc;
    __syncthreads();
#pragma unroll
    for (int off = THREADS / 2; off > 0; off >>= 1) {
        if (tid < off) red[tid] += red[tid + off];
        __syncthreads();
    }
    if (tid == 0) partial[blockIdx.x] = red[0];
}

// ---------------------------------------------------------------------------
// Final deterministic reduction of per-block partials -> scaled scalar loss.
// ---------------------------------------------------------------------------
__global__ __launch_bounds__(THREADS) void cel_var_reduce_kernel(
    const float* __restrict__ partial, int n, float* __restrict__ out, float scale)
{
    __shared__ float red[THREADS];
    float a = 0.0f;
    for (int i = threadIdx.x; i < n; i += THREADS) a += partial[i];  // fixed order
    red[threadIdx.x] = a;
    __syncthreads();
#pragma unroll
    for (int off = THREADS / 2; off > 0; off >>= 1) {
        if (threadIdx.x < off) red[threadIdx.x] += red[threadIdx.x + off];
        __syncthreads();
    }
    if (threadIdx.x == 0) out[0] = red[0] * scale;
}

extern "C" void kernel_launch(void* const* d_in, const int* in_sizes, int n_in,
                              void* d_out, int out_size, void* d_ws, size_t ws_size,
                              hipStream_t stream)
{
    (void)in_sizes; (void)n_in; (void)out_size; (void)ws_size;

    const float* mean  = (const float*)d_in[0];
    const float* logv  = (const float*)d_in[1];
    const int*   label = (const int*)d_in[2];
    const float* eps   = (const float*)d_in[3];
    float*       out   = (float*)d_out;
    float*       ws    = (float*)d_ws;   // >= 4096 floats of scratch

    const int npix    = B_DIM * HW_DIM;          // 1,048,576
    const int nblocks = npix / THREADS;          // 4096 (exact)

    cel_var_main_kernel<<<nblocks, THREADS, 0, stream>>>(mean, logv, label, eps, ws);

    const float scale = 1.0f / ((float)S_SAMP * (float)B_DIM * (float)HW_DIM);
    cel_var_reduce_kernel<<<1, THREADS, 0, stream>>>(ws, nblocks, out, scale);
}